// NMS_37924561224206
// MI455X (gfx1250) — compile-verified
//
#include <hip/hip_runtime.h>
#include <stdint.h>

#ifndef __has_builtin
#define __has_builtin(x) 0
#endif

// Problem constants (match reference setup_inputs / reference()).
#define NMS_B      16
#define NMS_N      131072
#define NMS_DET    100
#define NMS_THR    0.5f
#define NMS_TPB    1024
#define NMS_WAVES  (NMS_TPB / 32)          // 32 waves (wave32)
#define NMS_G      (NMS_N / 4 / NMS_TPB)   // 32 float4-groups per thread (128 elems)

typedef float v4f __attribute__((ext_vector_type(4)));

#if __has_builtin(__builtin_amdgcn_global_load_async_to_lds_b8)
#define NMS_HAS_ASYNC 1
typedef __attribute__((address_space(1))) char* nms_gptr;   // global, non-const char*
typedef __attribute__((address_space(3))) char* nms_lptr;   // LDS char*
#else
#define NMS_HAS_ASYNC 0
#endif

// wave32 max-reduction of a packed (score,index) u64 key
__device__ __forceinline__ unsigned long long wave_max_u64(unsigned long long k) {
#pragma unroll
  for (int off = 16; off > 0; off >>= 1) {
    unsigned long long o = __shfl_xor(k, off, 32);
    k = (o > k) ? o : k;
  }
  return k;
}

__global__ __launch_bounds__(NMS_TPB, 1)
void nms_greedy_kernel(const float* __restrict__ g_scores,
                       const v4f*  __restrict__ g_boxes,
                       const int*  __restrict__ g_classes,
                       int* __restrict__ g_out)
{
  // 128 KB class-byte cache: only legal because gfx1250 WGP has 320 KB LDS.
  __shared__ unsigned           s_clsw[NMS_N / 4];
  __shared__ unsigned long long s_wred[NMS_WAVES];
  __shared__ unsigned long long s_key;
  __shared__ float              s_sbox[5];   // x1,y1,x2,y2,area of selected box
  __shared__ int                s_scls;
  __shared__ int                s_sidx;

  const int b   = blockIdx.x;
  const int tid = threadIdx.x;

  const float* sc  = g_scores + (size_t)b * NMS_N;
  const v4f*   sc4 = (const v4f*)sc;
  const v4f*   bx  = g_boxes  + (size_t)b * NMS_N;
  const int*   cl  = g_classes + (size_t)b * NMS_N;
  int*         out = g_out + b * NMS_DET;

  // Per-thread base pointer so each unrolled score load folds j into the
  // 24-bit signed instruction offset (one address setup, 32 offset loads).
  const char* scb = (const char*)(sc4 + tid);

#if __has_builtin(__builtin_amdgcn_s_wait_tensorcnt)
  __builtin_amdgcn_s_wait_tensorcnt(0);   // CDNA5 tensor-counter path (no-op: count==0)
#endif

  // ---- Stage classes (0..79) into LDS as one byte per element ----
#if NMS_HAS_ASYNC
  {
    // Async DMA does the int32 -> u8 packing for free: each lane copies the
    // little-endian low byte of its class dword straight into LDS byte e.
    nms_gptr gsrc = (nms_gptr)cl;              // strips const; DMA only reads
    nms_lptr ldst = (nms_lptr)(char*)s_clsw;
#pragma unroll 8
    for (int r = 0; r < NMS_N / NMS_TPB; ++r) {       // 128 async_b8 per wave
      const int e = r * NMS_TPB + tid;
      __builtin_amdgcn_global_load_async_to_lds_b8(gsrc + 4 * e, ldst + e, 0, 0);
    }
  }
#if __has_builtin(__builtin_amdgcn_s_wait_asynccnt)
  __builtin_amdgcn_s_wait_asynccnt(0);
#else
  asm volatile("s_wait_asynccnt 0x0" ::: "memory");
#endif
#else
  {
    const int4* cl4 = (const int4*)cl;
#pragma unroll 4
    for (int g = tid; g < NMS_N / 4; g += NMS_TPB) {
      int4 c = cl4[g];
      s_clsw[g] = (unsigned)(c.x & 0xFF)         | ((unsigned)(c.y & 0xFF) << 8) |
                  ((unsigned)(c.z & 0xFF) << 16) | ((unsigned)(c.w & 0xFF) << 24);
    }
  }
#endif

  // Per-thread 128-bit validity mask in 4 VGPRs.
  // bit (j*4 + k)  <->  element  4*(j*1024 + tid) + k
  unsigned vm[4] = {~0u, ~0u, ~0u, ~0u};

  // Broadcast state of the most recent selection (none yet: class -1 matches nothing).
  float sx1 = 0.f, sy1 = 0.f, sx2 = 0.f, sy2 = 0.f, sarea = 0.f;
  int scls = -1;

  __syncthreads();

  for (int det = 0; det < NMS_DET; ++det) {
    // ---- fused pass: suppress vs previous selection + argmax for this round ----
    unsigned long long key = 0ULL;

#pragma unroll
    for (int j = 0; j < NMS_G; ++j) {
      const int w  = j >> 3;
      const int sh = (j & 7) * 4;
      unsigned nib = (vm[w] >> sh) & 0xFu;
      if (nib) {
        const int g = j * NMS_TPB + tid;
        if (j + 2 < NMS_G)   // hide L2 latency of the next tile (global_prefetch_b8)
          __builtin_prefetch((const void*)(scb + (size_t)(j + 2) * (NMS_TPB * 16)), 0, 3);
        const unsigned cw = s_clsw[g];                                  // ds_load_b32
        const v4f sv = *(const v4f*)(scb + (size_t)j * (NMS_TPB * 16)); // b128 @ imm offset
#pragma unroll
        for (int k = 0; k < 4; ++k) {
          const unsigned bit = 1u << k;
          if (nib & bit) {
            const int c = (int)((cw >> (8 * k)) & 0xFFu);
            if (c == scls) {             // only same-class boxes can be suppressed
              const int e = 4 * g + k;
              const v4f bb = bx[e];
              const float ix1 = fmaxf(sx1, bb.x);
              const float iy1 = fmaxf(sy1, bb.y);
              const float ix2 = fminf(sx2, bb.z);
              const float iy2 = fminf(sy2, bb.w);
              const float inter = fmaxf(ix2 - ix1, 0.f) * fmaxf(iy2 - iy1, 0.f);
              const float uni   = sarea + (bb.z - bb.x) * (bb.w - bb.y) - inter;
              if (uni > 0.f && (inter / uni) > NMS_THR) nib &= ~bit;
            }
            if (nib & bit) {
              // scores are in [0,1): positive-float bits are order-isomorphic.
              // Low word 0xFFFFFFFF-e gives JAX argmax first-occurrence tie-break.
              const unsigned e = 4u * (unsigned)g + (unsigned)k;
              const unsigned long long kk =
                  ((unsigned long long)__float_as_uint(sv[k]) << 32) |
                  (unsigned long long)(0xFFFFFFFFu - e);
              key = (kk > key) ? kk : key;
            }
          }
        }
        vm[w] = (vm[w] & ~(0xFu << sh)) | (nib << sh);
      }
    }

    // ---- block-wide argmax: wave32 shuffle tree + LDS across 32 waves ----
    const unsigned long long wk = wave_max_u64(key);
    if ((tid & 31) == 0) s_wred[tid >> 5] = wk;
    __syncthreads();
    if (tid < 32) {
      unsigned long long v = wave_max_u64(s_wred[tid]);
      if (tid == 0) {
        s_key = v;
        if (v != 0ULL) {
          const int e = (int)(0xFFFFFFFFu - (unsigned)(v & 0xFFFFFFFFull));
          const v4f sb = bx[e];
          s_sbox[0] = sb.x; s_sbox[1] = sb.y; s_sbox[2] = sb.z; s_sbox[3] = sb.w;
          s_sbox[4] = (sb.z - sb.x) * (sb.w - sb.y);
          s_scls = (int)((s_clsw[e >> 2] >> ((e & 3) * 8)) & 0xFFu);
          s_sidx = e;
        } else {
          s_sidx = -1;
        }
      }
    }
    __syncthreads();

    if (s_key == 0ULL) {                 // exhausted: pad remaining slots with -1
      if (tid == 0)
        for (int d = det; d < NMS_DET; ++d) out[d] = -1;
      break;
    }

    const int sel = s_sidx;
    if (tid == 0) out[det] = sel;

    // broadcast selected box/class into registers for the next fused pass
    sx1 = s_sbox[0]; sy1 = s_sbox[1]; sx2 = s_sbox[2]; sy2 = s_sbox[3];
    sarea = s_sbox[4];
    scls  = s_scls;

    // owner thread retires the selected element's validity bit (col == idx)
    const int g_sel = sel >> 2;
    if ((g_sel & (NMS_TPB - 1)) == tid) {
      const int jj = g_sel >> 10;
      const unsigned clr = ~(1u << (((jj & 7) * 4) + (sel & 3)));
      const int ww = jj >> 3;
      if      (ww == 0) vm[0] &= clr;
      else if (ww == 1) vm[1] &= clr;
      else if (ww == 2) vm[2] &= clr;
      else              vm[3] &= clr;
    }
  }
}

extern "C" void kernel_launch(void* const* d_in, const int* in_sizes, int n_in,
                              void* d_out, int out_size, void* d_ws, size_t ws_size,
                              hipStream_t stream) {
  (void)in_sizes; (void)n_in; (void)d_ws; (void)ws_size; (void)out_size;
  const float* scores  = (const float*)d_in[0];
  const v4f*   boxes   = (const v4f*)d_in[1];
  const int*   classes = (const int*)d_in[2];
  int*         out     = (int*)d_out;

  nms_greedy_kernel<<<dim3(NMS_B), dim3(NMS_TPB), 0, stream>>>(scores, boxes, classes, out);
}